// CovCorrLog_22531398434820
// MI455X (gfx1250) — compile-verified
//
#include <hip/hip_runtime.h>
#include <math.h>

// MPNCOV covariance pooling -> NS matrix sqrt -> correlation -> logm -> triuvec
// logm is computed with inverse scaling-and-squaring (5 NS square roots +
// degree-8 Mercator series), i.e. pure 256^3 batched GEMMs on the f32 WMMA pipe.
//
// Workspace layout (floats):
//   S0..S5 : 6 x (128*256*256)  square matrix slots   (~201.3 MB)
//   meanb  : 128*256            per-row spatial means
//   trb    : 128                traces
//   rsb    : 128                sqrt(traces)

#define BATCH   128
#define CDIM    256
#define MDIM    784          // 28*28
#define ITER_N  5            // MPNCOV Newton-Schulz iterations
#define K_ROOTS 5            // inverse scaling-squaring depth for logm
#define OUT_PER_B ((CDIM * (CDIM + 1)) / 2)   // 32896

typedef float v2f __attribute__((ext_vector_type(2)));
typedef float v8f __attribute__((ext_vector_type(8)));

// ---------------- per-row spatial mean ----------------
__global__ __launch_bounds__(256)
void mpn_row_mean(const float* __restrict__ x, float* __restrict__ mean) {
  const int row = blockIdx.x;                      // b*C + c
  const float* src = x + (size_t)row * MDIM;
  float s = 0.f;
  for (int i = threadIdx.x; i < MDIM; i += 256) s += src[i];
  __shared__ float red[256];
  red[threadIdx.x] = s;
  __syncthreads();
  for (int off = 128; off > 0; off >>= 1) {
    if (threadIdx.x < off) red[threadIdx.x] += red[threadIdx.x + off];
    __syncthreads();
  }
  if (threadIdx.x == 0) mean[row] = red[0] * (1.0f / MDIM);
}

// ---------------- covariance: cov = (x-mu)(x-mu)^T / M ----------------
// 128x128 output tile per block, K chunks of 56, centering fused into loads.
__global__ __launch_bounds__(256)
void mpn_cov_gemm(const float* __restrict__ x, const float* __restrict__ mean,
                  float* __restrict__ D) {
  __shared__ float Ash[128][57];   // odd stride -> conflict-free frag reads
  __shared__ float Bsh[56][129];
  const int b = blockIdx.z;
  const int m_base = blockIdx.y * 128;
  const int n_base = blockIdx.x * 128;
  const float* xb = x + (size_t)b * CDIM * MDIM;
  const float* mb = mean + b * CDIM;
  float* Db = D + (size_t)b * CDIM * CDIM;

  const int t = threadIdx.x;
  const int wid = t >> 5, lane = t & 31, lo = lane & 15, hi = lane >> 4;
  const int wrow = (wid >> 1) * 32, wcol = (wid & 1) * 64;

  const v8f vzero = {0.f, 0.f, 0.f, 0.f, 0.f, 0.f, 0.f, 0.f};
  v8f acc[2][4];
#pragma unroll
  for (int mi = 0; mi < 2; ++mi)
#pragma unroll
    for (int ni = 0; ni < 4; ++ni) acc[mi][ni] = vzero;

  for (int kk = 0; kk < MDIM; kk += 56) {
    {
      const int row = t >> 1, half = t & 1;   // 2 threads per row, 28 floats each
      const float mu = mb[m_base + row];
      const float* src = xb + (size_t)(m_base + row) * MDIM + kk + half * 28;
      float* dst = &Ash[row][half * 28];
#pragma unroll
      for (int i = 0; i < 7; ++i) {
        float4 v = *(const float4*)(src + 4 * i);
        dst[4 * i + 0] = v.x - mu; dst[4 * i + 1] = v.y - mu;
        dst[4 * i + 2] = v.z - mu; dst[4 * i + 3] = v.w - mu;
      }
      const float mun = mb[n_base + row];
      const float* srcb = xb + (size_t)(n_base + row) * MDIM + kk + half * 28;
#pragma unroll
      for (int i = 0; i < 7; ++i) {
        float4 v = *(const float4*)(srcb + 4 * i);
        Bsh[half * 28 + 4 * i + 0][row] = v.x - mun;
        Bsh[half * 28 + 4 * i + 1][row] = v.y - mun;
        Bsh[half * 28 + 4 * i + 2][row] = v.z - mun;
        Bsh[half * 28 + 4 * i + 3][row] = v.w - mun;
      }
      if (kk + 56 < MDIM) {
        __builtin_prefetch(src + 56, 0, 1);
        __builtin_prefetch(srcb + 56, 0, 1);
      }
    }
    __syncthreads();
#pragma unroll
    for (int ks = 0; ks < 14; ++ks) {
      const int k0 = ks * 4 + 2 * hi;          // lanes 16-31 hold K=2,3
      v2f af[2], bf[4];
#pragma unroll
      for (int mi = 0; mi < 2; ++mi) {
        const int m = wrow + mi * 16 + lo;
        af[mi].x = Ash[m][k0]; af[mi].y = Ash[m][k0 + 1];
      }
#pragma unroll
      for (int ni = 0; ni < 4; ++ni) {
        const int n = wcol + ni * 16 + lo;
        bf[ni].x = Bsh[k0][n]; bf[ni].y = Bsh[k0 + 1][n];
      }
#pragma unroll
      for (int mi = 0; mi < 2; ++mi)
#pragma unroll
        for (int ni = 0; ni < 4; ++ni)
          acc[mi][ni] = __builtin_amdgcn_wmma_f32_16x16x4_f32(
              false, af[mi], false, bf[ni], (short)0, acc[mi][ni], false, false);
    }
    __syncthreads();
  }

  const float s = 1.0f / (float)MDIM;
#pragma unroll
  for (int mi = 0; mi < 2; ++mi)
#pragma unroll
    for (int ni = 0; ni < 4; ++ni) {
      const int n = n_base + wcol + ni * 16 + lo;
#pragma unroll
      for (int v = 0; v < 8; ++v) {
        const int m = m_base + wrow + mi * 16 + 8 * hi + v;  // ISA C/D layout
        Db[(size_t)m * CDIM + n] = s * acc[mi][ni][v];
      }
    }
}

// ---------------- batched 256x256x256 GEMM:  D = scale*(bscale[b])*(A@B) + diag*I
// Block (256 thr = 8 waves) computes 128(m) x 256(n); each wave owns a 64x64
// tile = 4x4 WMMA subtiles -> 16 v_wmma per k-step against 8 LDS pair-loads.
__global__ __launch_bounds__(256)
void mpn_gemm256(const float* __restrict__ A, const float* __restrict__ Bm,
                 float* __restrict__ D, float scale, float diag_add,
                 const float* __restrict__ bscale) {
  __shared__ float Ash[128][33];
  __shared__ float Bsh[32][260];
  const int b = blockIdx.z;
  const int m_base = blockIdx.x * 128;   // grid.x = 2
  const float* Ab = A + (size_t)b * CDIM * CDIM;
  const float* Bb = Bm + (size_t)b * CDIM * CDIM;
  float* Db = D + (size_t)b * CDIM * CDIM;

  const int t = threadIdx.x;
  const int wid = t >> 5, lane = t & 31, lo = lane & 15, hi = lane >> 4;
  const int wrow = (wid >> 2) * 64;      // 0 or 64   (m within block tile)
  const int wcol = (wid & 3) * 64;       // 0,64,128,192 (n)

  const v8f vzero = {0.f, 0.f, 0.f, 0.f, 0.f, 0.f, 0.f, 0.f};
  v8f acc[4][4];
#pragma unroll
  for (int mi = 0; mi < 4; ++mi)
#pragma unroll
    for (int ni = 0; ni < 4; ++ni) acc[mi][ni] = vzero;

  for (int kk = 0; kk < CDIM; kk += 32) {
    {
      const int row = t >> 1, half = t & 1;          // A tile 128x32
      const float* src = Ab + (size_t)(m_base + row) * CDIM + kk + half * 16;
      float* dst = &Ash[row][half * 16];
#pragma unroll
      for (int i = 0; i < 4; ++i) {
        float4 v = *(const float4*)(src + 4 * i);
        dst[4 * i + 0] = v.x; dst[4 * i + 1] = v.y;
        dst[4 * i + 2] = v.z; dst[4 * i + 3] = v.w;
      }
      const int brow = t >> 3, q = t & 7;            // B tile 32x256
      const float* srcb = Bb + (size_t)(kk + brow) * CDIM + q * 32;
      float* dstb = &Bsh[brow][q * 32];
#pragma unroll
      for (int i = 0; i < 8; ++i) {
        float4 v = *(const float4*)(srcb + 4 * i);
        dstb[4 * i + 0] = v.x; dstb[4 * i + 1] = v.y;
        dstb[4 * i + 2] = v.z; dstb[4 * i + 3] = v.w;
      }
      if (kk + 32 < CDIM) {                          // next-chunk prefetch
        __builtin_prefetch(src + 32, 0, 1);
        __builtin_prefetch(srcb + 32 * CDIM, 0, 1);
      }
    }
    __syncthreads();
#pragma unroll
    for (int ks = 0; ks < 8; ++ks) {
      const int k0 = ks * 4 + 2 * hi;
      v2f af[4], bf[4];
#pragma unroll
      for (int mi = 0; mi < 4; ++mi) {
        const int m = wrow + mi * 16 + lo;
        af[mi].x = Ash[m][k0]; af[mi].y = Ash[m][k0 + 1];
      }
#pragma unroll
      for (int ni = 0; ni < 4; ++ni) {
        const int n = wcol + ni * 16 + lo;
        bf[ni].x = Bsh[k0][n]; bf[ni].y = Bsh[k0 + 1][n];
      }
#pragma unroll
      for (int mi = 0; mi < 4; ++mi)
#pragma unroll
        for (int ni = 0; ni < 4; ++ni)
          acc[mi][ni] = __builtin_amdgcn_wmma_f32_16x16x4_f32(
              false, af[mi], false, bf[ni], (short)0, acc[mi][ni], false, false);
    }
    __syncthreads();
  }

  const float s = scale * (bscale ? bscale[b] : 1.0f);
#pragma unroll
  for (int mi = 0; mi < 4; ++mi)
#pragma unroll
    for (int ni = 0; ni < 4; ++ni) {
      const int n = wcol + ni * 16 + lo;
#pragma unroll
      for (int v = 0; v < 8; ++v) {
        const int m = m_base + wrow + mi * 16 + 8 * hi + v;
        float val = s * acc[mi][ni][v];
        if (m == n) val += diag_add;
        Db[(size_t)m * CDIM + n] = val;
      }
    }
}

// ---------------- trace + sqrt(trace) per batch ----------------
__global__ __launch_bounds__(256)
void mpn_trace(const float* __restrict__ X, float* __restrict__ tr,
               float* __restrict__ rs) {
  const int b = blockIdx.x;
  const int i = threadIdx.x;
  __shared__ float red[256];
  red[i] = X[(size_t)b * CDIM * CDIM + (size_t)i * CDIM + i];
  __syncthreads();
  for (int off = 128; off > 0; off >>= 1) {
    if (i < off) red[i] += red[i + off];
    __syncthreads();
  }
  if (i == 0) { tr[b] = red[0]; rs[b] = sqrtf(red[0]); }
}

// ---------------- NS init: An = X/tr ; Z0 = 1.5 I - 0.5 An ----------------
__global__ __launch_bounds__(256)
void mpn_ns_init(const float* __restrict__ X, const float* __restrict__ tr,
                 float* __restrict__ An, float* __restrict__ Z0) {
  const int b = blockIdx.y;
  const int idx = blockIdx.x * 256 + threadIdx.x;  // 0..65535
  const int i = idx >> 8, j = idx & 255;
  const size_t o = (size_t)b * CDIM * CDIM + idx;
  const float a = X[o] / tr[b];
  An[o] = a;
  Z0[o] = (i == j ? 1.5f : 0.0f) - 0.5f * a;
}

// ---------------- correlation normalization ----------------
__global__ __launch_bounds__(256)
void mpn_corr(const float* __restrict__ S, float* __restrict__ Cc) {
  const int b = blockIdx.y;
  const int idx = blockIdx.x * 256 + threadIdx.x;
  const int i = idx >> 8, j = idx & 255;
  const size_t base = (size_t)b * CDIM * CDIM;
  const float eps = 1.1920929e-07f;
  const float di = S[base + (size_t)i * CDIM + i];
  const float dj = S[base + (size_t)j * CDIM + j];
  float norm = sqrtf(fmaxf(di, eps)) * sqrtf(fmaxf(dj, eps));
  if (norm == 0.0f) norm = 1.0f;
  Cc[base + idx] = S[base + idx] / norm;
}

// ---------------- out = a*in + bdiag*I ----------------
__global__ __launch_bounds__(256)
void mpn_axpbi(const float* __restrict__ in, float* __restrict__ out,
               float a, float bdiag) {
  const int b = blockIdx.y;
  const int idx = blockIdx.x * 256 + threadIdx.x;
  const int i = idx >> 8, j = idx & 255;
  const size_t o = (size_t)b * CDIM * CDIM + idx;
  out[o] = a * in[o] + (i == j ? bdiag : 0.0f);
}

// ---------------- OLM + triuvec fused ----------------
__device__ __forceinline__ int triu_off(int r) {
  return r * CDIM - (r * (r - 1)) / 2;
}

__global__ __launch_bounds__(256)
void mpn_triu_olm(const float* __restrict__ L, float* __restrict__ out) {
  const int b = blockIdx.y;
  const int t = blockIdx.x * 256 + threadIdx.x;
  if (t >= OUT_PER_B) return;
  const double d = 2.0 * CDIM + 1.0;
  int r = (int)((d - sqrt(d * d - 8.0 * (double)t)) * 0.5);
  if (r < 0) r = 0;
  if (r > CDIM - 1) r = CDIM - 1;
  while (r > 0 && triu_off(r) > t) --r;
  while (r < CDIM - 1 && triu_off(r + 1) <= t) ++r;
  const int c = r + (t - triu_off(r));
  const size_t base = (size_t)b * CDIM * CDIM;
  float v = 0.0f;
  if (r != c)
    v = L[base + (size_t)r * CDIM + c] + L[base + (size_t)c * CDIM + r];
  out[(size_t)b * OUT_PER_B + t] = v;
}

// ---------------- host orchestration ----------------
extern "C" void kernel_launch(void* const* d_in, const int* in_sizes, int n_in,
                              void* d_out, int out_size, void* d_ws, size_t ws_size,
                              hipStream_t stream) {
  (void)in_sizes; (void)n_in; (void)out_size; (void)ws_size;
  const float* x = (const float*)d_in[0];
  float* out = (float*)d_out;

  const size_t MATB = (size_t)BATCH * CDIM * CDIM;
  float* S0 = (float*)d_ws;
  float* S1 = S0 + MATB;
  float* S2 = S1 + MATB;
  float* S3 = S2 + MATB;
  float* S4 = S3 + MATB;
  float* S5 = S4 + MATB;
  float* meanb = S5 + MATB;          // BATCH*CDIM
  float* trb   = meanb + BATCH * CDIM;
  float* rsb   = trb + BATCH;

  auto gemm = [&](const float* A, const float* B, float* D, float sc, float dg,
                  const float* bs) {
    mpn_gemm256<<<dim3(2, 1, BATCH), 256, 0, stream>>>(A, B, D, sc, dg, bs);
  };

  // Trace-normalized Newton-Schulz sqrt with sqrt(trace) rescale.
  // OUT may alias X. anT serves as An first, then T (An is dead after Y0).
  auto ns_sqrt = [&](float* X, float* OUT, float* anT, float* z0, float* z1,
                     float* y0, float* y1) {
    mpn_trace<<<BATCH, 256, 0, stream>>>(X, trb, rsb);
    mpn_ns_init<<<dim3(256, BATCH), 256, 0, stream>>>(X, trb, anT, z0);
    gemm(anT, z0, y0, 1.0f, 0.0f, nullptr);          // Y0 = An @ Z0
    float *Yc = y0, *Yn = y1, *Zc = z0, *Zn = z1;
    for (int it = 0; it < ITER_N - 2; ++it) {
      gemm(Zc, Yc, anT, -0.5f, 1.5f, nullptr);       // T = 1.5I - 0.5 Z@Y
      gemm(Yc, anT, Yn, 1.0f, 0.0f, nullptr);        // Y' = Y@T
      gemm(anT, Zc, Zn, 1.0f, 0.0f, nullptr);        // Z' = T@Z
      float* tp = Yc; Yc = Yn; Yn = tp;
      tp = Zc; Zc = Zn; Zn = tp;
    }
    gemm(Zc, Yc, anT, -0.5f, 1.5f, nullptr);
    gemm(Yc, anT, OUT, 1.0f, 0.0f, rsb);             // rescale by sqrt(tr)
  };

  // 1) covariance pooling
  mpn_row_mean<<<BATCH * CDIM, 256, 0, stream>>>(x, meanb);
  mpn_cov_gemm<<<dim3(2, 2, BATCH), 256, 0, stream>>>(x, meanb, S0);

  // 2) MPNCOV matrix sqrt (rescale cancels inside correlation)
  ns_sqrt(S0, S0, S1, S2, S3, S4, S5);

  // 3) correlation normalization
  mpn_corr<<<dim3(256, BATCH), 256, 0, stream>>>(S0, S1);

  // 4) logm via inverse scaling-and-squaring: k repeated sqrts ...
  for (int r = 0; r < K_ROOTS; ++r)
    ns_sqrt(S1, S1, S0, S2, S3, S4, S5);

  // ... then log(I+E) Mercator series, degree 8, Horner in GEMMs.
  mpn_axpbi<<<dim3(256, BATCH), 256, 0, stream>>>(S1, S0, 1.0f, -1.0f);  // E = X - I
  mpn_axpbi<<<dim3(256, BATCH), 256, 0, stream>>>(S0, S2, -1.0f / 8.0f,
                                                  1.0f / 7.0f);          // G7 = a8 E + a7 I
  float *Gc = S2, *Gn = S3;
  for (int m = 6; m >= 1; --m) {
    const float am = ((m & 1) ? 1.0f : -1.0f) / (float)m;
    gemm(S0, Gc, Gn, 1.0f, am, nullptr);             // G = am I + E@G
    float* tp = Gc; Gc = Gn; Gn = tp;
  }
  gemm(S0, Gc, S4, (float)(1 << K_ROOTS), 0.0f, nullptr);  // logC = 2^k * E@G

  // 5) OLM + triuvec
  mpn_triu_olm<<<dim3((OUT_PER_B + 255) / 256, BATCH), 256, 0, stream>>>(S4, out);
}